// Sinkhorn_48180943126738
// MI455X (gfx1250) — compile-verified
//
#include <hip/hip_runtime.h>
#include <math.h>

// ---------------- problem constants ----------------
static constexpr int    NROWS = 1500;          // points per cloud
static constexpr int    NP    = 1504;          // padded to 16
static constexpr int    DD    = 300;           // feature dim (= 4*75, exact for K=4 WMMA)
static constexpr int    NB    = 4;             // batch pairs
static constexpr int    NTILE = NP / 16;       // 94 tiles per dim
static constexpr int    NITER = 30;

static constexpr float  EPSV    = 2.5e-9f;     // blur^2
static constexpr float  LOG2E_F = 1.44269504088896340736f;
static constexpr float  LN2_F   = 0.69314718055994530942f;
// All log-domain quantities are stored in base-2 units: S2 = -C/(eps*ln2).
// This makes every exp in the hot loop a raw exp2 (no per-element LOG2E multiply).
static constexpr float  S_SCALE = 4.0e8f * LOG2E_F;   // (1/eps) * log2(e)

// ---------------- workspace layout (float offsets) ----------------
static constexpr size_t PXSZ   = (size_t)NB * NP * DD;
static constexpr size_t PX_OFF = 0;
static constexpr size_t PY_OFF = PX_OFF + PXSZ;
static constexpr size_t NXSZ   = (size_t)NB * NP;
static constexpr size_t NX_OFF = PY_OFF + PXSZ;
static constexpr size_t NY_OFF = NX_OFF + NXSZ;
static constexpr size_t SSZ    = (size_t)NP * NP;           // one 1504x1504 matrix
static constexpr size_t S_OFF  = NY_OFF + NXSZ;             // 16 matrices: Sxy[4], Syx[4], Sself[8]
static constexpr size_t POT_OFF = S_OFF + 16 * SSZ;
static constexpr size_t F_OFF  = POT_OFF;                   // F[4][NP]
static constexpr size_t G_OFF  = POT_OFF + (size_t)4 * NP;  // G[4][NP]
static constexpr size_t P0_OFF = POT_OFF + (size_t)8 * NP;  // P buf0 [8][NP]
static constexpr size_t P1_OFF = POT_OFF + (size_t)16 * NP; // P buf1 [8][NP]
static constexpr size_t POT_TOTAL = (size_t)24 * NP;

typedef __attribute__((ext_vector_type(2))) float v2f;
typedef __attribute__((ext_vector_type(8))) float v8f;

// ---------------- zero potentials ----------------
__global__ void sk_init_pots(float* __restrict__ ws) {
    size_t i = (size_t)blockIdx.x * blockDim.x + threadIdx.x;
    if (i < POT_TOTAL) ws[POT_OFF + i] = 0.0f;
}

// ---------------- px = x^2 / colsum(x^2), zero-padded to NP rows ----------------
__global__ void sk_normalize(const float* __restrict__ x, const float* __restrict__ y,
                             float* __restrict__ ws) {
    const int s = blockIdx.y;              // 0..7 : {x b0..3, y b0..3}
    const int b = s & 3;
    const float* in = (s < 4 ? x : y) + (size_t)b * NROWS * DD;
    float* out = ws + (s < 4 ? PX_OFF : PY_OFF) + (size_t)b * NP * DD;
    __shared__ float inv[DD];
    for (int d = threadIdx.x; d < DD; d += blockDim.x) {
        float acc = 0.0f;
        for (int i = 0; i < NROWS; ++i) {
            float v = in[(size_t)i * DD + d];
            acc += v * v;
        }
        inv[d] = 1.0f / acc;
    }
    __syncthreads();
    for (size_t idx = threadIdx.x; idx < (size_t)NP * DD; idx += blockDim.x) {
        int i = (int)(idx / DD), d = (int)(idx % DD);
        float v = 0.0f;
        if (i < NROWS) {
            float t = in[(size_t)i * DD + d];
            v = t * t * inv[d];
        }
        out[idx] = v;
    }
}

// ---------------- row squared norms (wave per row, wave32 shuffle reduce) ----------------
__global__ __launch_bounds__(256) void sk_rownorm(float* __restrict__ ws) {
    const int s = blockIdx.y;              // 0..7
    const int b = s & 3;
    const float* base = ws + (s < 4 ? PX_OFF : PY_OFF) + (size_t)b * NP * DD;
    float* out = ws + (s < 4 ? NX_OFF : NY_OFF) + (size_t)b * NP;
    const int wave = threadIdx.x >> 5, lane = threadIdx.x & 31;
    const int row = blockIdx.x * 8 + wave;           // gridDim.x = 188 -> 1504 rows
    const float* rp = base + (size_t)row * DD;
    float acc = 0.0f;
    for (int j = lane * 4; j + 4 <= DD; j += 128) {
        float4 v = *(const float4*)(rp + j);
        acc += v.x * v.x + v.y * v.y + v.z * v.z + v.w * v.w;
    }
    for (int off = 16; off; off >>= 1) acc += __shfl_xor(acc, off, 32);
    if (lane == 0) out[row] = acc;
}

// ---------------- WMMA cost GEMM: S2 = (px.py^T - 0.5*(nx_i+ny_j)) * (log2e/eps) -------
// One wave per 16x16 output tile; fp32 V_WMMA_F32_16X16X4_F32, K-loop of 75 steps.
// For the xy set also writes the transpose matrix (contiguous float4 per lane).
__global__ __launch_bounds__(128) void sk_cost_gemm(float* __restrict__ ws) {
    const int s = blockIdx.y;              // 0..11
    size_t aoff, boff, naoff, nboff, ooff;
    long long toff = -1;
    if (s < 4) {
        const int b = s;
        aoff = PX_OFF + (size_t)b * NP * DD;  boff = PY_OFF + (size_t)b * NP * DD;
        naoff = NX_OFF + (size_t)b * NP;      nboff = NY_OFF + (size_t)b * NP;
        ooff = S_OFF + (size_t)b * SSZ;       toff = (long long)(S_OFF + (size_t)(4 + b) * SSZ);
    } else if (s < 8) {
        const int b = s - 4;
        aoff = boff = PX_OFF + (size_t)b * NP * DD;
        naoff = nboff = NX_OFF + (size_t)b * NP;
        ooff = S_OFF + (size_t)(8 + 2 * b) * SSZ;
    } else {
        const int b = s - 8;
        aoff = boff = PY_OFF + (size_t)b * NP * DD;
        naoff = nboff = NY_OFF + (size_t)b * NP;
        ooff = S_OFF + (size_t)(8 + 2 * b + 1) * SSZ;
    }
    const int wave = threadIdx.x >> 5, lane = threadIdx.x & 31;
    const int t  = blockIdx.x * 4 + wave;          // gridDim.x = 2209 -> 8836 tiles
    const int ti = t / NTILE, tj = t % NTILE;
    const int lo = lane & 15, hi = lane >> 4;

    // A frag: lane holds row i0+lo, K pair (k+2*hi, k+1+2*hi). B frag identical on py.
    const float* arow = ws + aoff + (size_t)(ti * 16 + lo) * DD + 2 * hi;
    const float* brow = ws + boff + (size_t)(tj * 16 + lo) * DD + 2 * hi;

    v8f acc = {0.f, 0.f, 0.f, 0.f, 0.f, 0.f, 0.f, 0.f};
#pragma unroll 5
    for (int k = 0; k < DD; k += 4) {
        v2f a = *(const v2f*)(arow + k);
        v2f bb = *(const v2f*)(brow + k);
        acc = __builtin_amdgcn_wmma_f32_16x16x4_f32(false, a, false, bb,
                                                    (short)0, acc, false, false);
    }

    // epilogue: rank-1 bias + scale; C layout lane L -> (M = r + 8*hi, N = lo).
    // nx values for this lane's 8 rows are contiguous & 16B aligned -> two float4 loads.
    const float  nyj = ws[nboff + (size_t)(tj * 16 + lo)];
    const float* nxp = ws + naoff + (size_t)(ti * 16 + 8 * hi);
    const float4 nxa = *(const float4*)(nxp);
    const float4 nxb = *(const float4*)(nxp + 4);
    const float  nxr[8] = {nxa.x, nxa.y, nxa.z, nxa.w, nxb.x, nxb.y, nxb.z, nxb.w};
    float* out = ws + ooff;
    const size_t col = (size_t)tj * 16 + lo;
    float sv[8];
#pragma unroll
    for (int r = 0; r < 8; ++r)
        sv[r] = (acc[r] - 0.5f * (nxr[r] + nyj)) * S_SCALE;
#pragma unroll
    for (int r = 0; r < 8; ++r)
        out[(size_t)(ti * 16 + 8 * hi + r) * NP + col] = sv[r];
    if (toff >= 0) {
        float* tp = ws + (size_t)toff + col * NP + (size_t)(ti * 16 + 8 * hi);
        *(float4*)(tp)     = make_float4(sv[0], sv[1], sv[2], sv[3]);
        *(float4*)(tp + 4) = make_float4(sv[4], sv[5], sv[6], sv[7]);
    }
}

// ---------------- softmin sweep (base-2 log units) -------------------------------------
// pout_i = -log2sumexp2_j(logw2 + pin_j + S2_ij)   (optional 0.5 mix for self potentials)
// Two set-groups per launch so independent updates share one kernel launch.
// 8 waves/block, wave per row, potentials staged in LDS, float4 loads, online LSE.
__global__ __launch_bounds__(256) void sk_softmin(
        const float* __restrict__ C1, const float* __restrict__ pin1,
        float* __restrict__ pout1, int mix1, int n1,
        const float* __restrict__ C2, const float* __restrict__ pin2,
        float* __restrict__ pout2, int mix2, float logw2) {
    __shared__ float sp[NP];
    const int s = blockIdx.y;
    const float* C;  const float* pin;  float* pout;  int mix;
    if (s < n1) { C = C1 + (size_t)s * SSZ;  pin = pin1 + (size_t)s * NP;
                  pout = pout1 + (size_t)s * NP;  mix = mix1; }
    else        { const int q = s - n1;
                  C = C2 + (size_t)q * SSZ;  pin = pin2 + (size_t)q * NP;
                  pout = pout2 + (size_t)q * NP;  mix = mix2; }
    for (int i = threadIdx.x; i < NP; i += 256) sp[i] = pin[i] + logw2;
    __syncthreads();

    const int wave = threadIdx.x >> 5, lane = threadIdx.x & 31;
    const int row = blockIdx.x * 8 + wave;
    if (row >= NROWS) return;
    const float* rp = C + (size_t)row * NP;

    float m = -INFINITY, ss = 0.0f;
    for (int j = lane * 4; j + 4 <= NROWS; j += 128) {     // 1500 % 4 == 0 -> exact cover
        float4 c = *(const float4*)(rp + j);
        float4 p = *(const float4*)(&sp[j]);
        float t0 = c.x + p.x, t1 = c.y + p.y, t2 = c.z + p.z, t3 = c.w + p.w;
        float mx = fmaxf(fmaxf(t0, t1), fmaxf(t2, t3));
        float nm = fmaxf(m, mx);
        ss = ss * exp2f(m - nm)
           + exp2f(t0 - nm) + exp2f(t1 - nm) + exp2f(t2 - nm) + exp2f(t3 - nm);
        m = nm;
    }
    for (int off = 16; off; off >>= 1) {
        float om = __shfl_xor(m, off, 32);
        float os = __shfl_xor(ss, off, 32);
        float nm = fmaxf(m, om);
        ss = ss * exp2f(m - nm) + os * exp2f(om - nm);
        m = nm;
    }
    if (lane == 0) {
        float F = -(m + log2f(ss));
        pout[row] = mix ? 0.5f * (pin[row] + F) : F;
    }
}

// ------- final scalar: B * sum_b eps*ln2*(mean F2 + mean G2 - mean Pxx2 - mean Pyy2) ----
__global__ void sk_finalize(const float* __restrict__ ws, float* __restrict__ out) {
    __shared__ float red[256];
    float acc = 0.0f;
    for (int idx = threadIdx.x; idx < NB * NROWS; idx += 256) {
        const int b = idx / NROWS, i = idx % NROWS;
        acc += ws[F_OFF + (size_t)b * NP + i] + ws[G_OFF + (size_t)b * NP + i]
             - ws[P0_OFF + (size_t)(2 * b) * NP + i]
             - ws[P0_OFF + (size_t)(2 * b + 1) * NP + i];
    }
    red[threadIdx.x] = acc;
    __syncthreads();
    for (int o = 128; o; o >>= 1) {
        if ((int)threadIdx.x < o) red[threadIdx.x] += red[threadIdx.x + o];
        __syncthreads();
    }
    if (threadIdx.x == 0) out[0] = 4.0f * EPSV * LN2_F * red[0] / 1500.0f;
}

extern "C" void kernel_launch(void* const* d_in, const int* in_sizes, int n_in,
                              void* d_out, int out_size, void* d_ws, size_t ws_size,
                              hipStream_t stream) {
    (void)in_sizes; (void)n_in; (void)out_size; (void)ws_size;
    const float* x = (const float*)d_in[0];
    const float* y = (const float*)d_in[1];
    float* ws  = (float*)d_ws;
    float* out = (float*)d_out;
    const float logw2 = -logf(1500.0f) * LOG2E_F;   // base-2 units

    sk_init_pots<<<(unsigned)((POT_TOTAL + 255) / 256), 256, 0, stream>>>(ws);
    sk_normalize<<<dim3(1, 8), 256, 0, stream>>>(x, y, ws);
    sk_rownorm<<<dim3(188, 8), 256, 0, stream>>>(ws);
    sk_cost_gemm<<<dim3(2209, 12), 128, 0, stream>>>(ws);

    float* Sxy = ws + S_OFF;
    float* Syx = ws + S_OFF + 4 * SSZ;
    float* Ssf = ws + S_OFF + 8 * SSZ;
    float* F = ws + F_OFF;  float* G = ws + G_OFF;

    for (int t = 0; t < NITER; ++t) {
        // f = softmin(Cxy, g)                     (4 sets)
        sk_softmin<<<dim3(188, 4), 256, 0, stream>>>(Sxy, G, F, 0, 4,
                                                     Sxy, G, F, 0, logw2);
        // g = softmin(Cxy^T, f)  fused with independent self-potential ping-pong (12 sets)
        float* pin  = ws + ((t & 1) ? P1_OFF : P0_OFF);
        float* pout = ws + ((t & 1) ? P0_OFF : P1_OFF);   // t=29 (odd) lands in P0
        sk_softmin<<<dim3(188, 12), 256, 0, stream>>>(Syx, F, G, 0, 4,
                                                      Ssf, pin, pout, 1, logw2);
    }
    sk_finalize<<<1, 256, 0, stream>>>(ws, out);
}